// ChamferLoss_65051574665232
// MI455X (gfx1250) — compile-verified
//
#include <hip/hip_runtime.h>

// CDNA5 / gfx1250 Chamfer distance using V_WMMA_F32_16X16X4_F32.
// d(p,g) = |p|^2 + |g|^2 - 2 p.g  computed as  WMMA( A=(-2p,1), B=(g,|g|^2) ) + |p|^2
// Min over columns is accumulated on raw WMMA output (monotone transform applied after).

typedef __attribute__((ext_vector_type(2))) float v2f;
typedef __attribute__((ext_vector_type(8))) float v8f;

#define WAVES_PER_BLOCK 8
#define ROWS_PER_WAVE   16
#define ROWS_PER_BLOCK  (WAVES_PER_BLOCK * ROWS_PER_WAVE)   // 128

// One wave owns 16 "row" points (A matrix, fixed for the whole kernel) and
// sweeps all "col" points 16 at a time with WMMA, keeping per-row running mins.
// partial[blockIdx.x] = sum over this block's 128 rows of min_j d(row_i, col_j).
__global__ __launch_bounds__(256) void chamfer_min_rows(
    const float* __restrict__ rows_pts,   // [B, R, 3]
    const float* __restrict__ cols_pts,   // [B, C, 3]
    int R, int C, int blocksPerBatch,
    float* __restrict__ partial)
{
    const int tid  = threadIdx.x;
    const int wave = tid >> 5;
    const int lane = tid & 31;
    const int half = lane >> 4;   // 0: K0/K1 half, 1: K2/K3 half
    const int l16  = lane & 15;

    const int batch   = blockIdx.x / blocksPerBatch;
    const int rowBase = (blockIdx.x % blocksPerBatch) * ROWS_PER_BLOCK + wave * ROWS_PER_WAVE;

    const float* P = rows_pts + (size_t)batch * (size_t)R * 3u;
    const float* G = cols_pts + (size_t)batch * (size_t)C * 3u;

    // ---- A operand: 16x4 f32. Row m = (-2px, -2py, -2pz, 1).
    // Layout (ISA 7.12.2): lanes 0-15 hold {K0,K1} for M=lane; lanes 16-31 hold {K2,K3}.
    v2f a;
    {
        const float* pp = P + (size_t)(rowBase + l16) * 3u;
        float px = pp[0], py = pp[1], pz = pp[2];
        a.x = half ? (-2.0f * pz) : (-2.0f * px);
        a.y = half ? 1.0f         : (-2.0f * py);
    }

    // |p|^2 for the 8 D-matrix rows this lane sees: rows rowBase + r + 8*half.
    float x2r[8];
#pragma unroll
    for (int r = 0; r < 8; ++r) {
        const float* pp = P + (size_t)(rowBase + r + 8 * half) * 3u;
        float qx = pp[0], qy = pp[1], qz = pp[2];
        x2r[r] = qx * qx + qy * qy + qz * qz;
    }

    const float INF = 3.0e38f;
    v8f mn = { INF, INF, INF, INF, INF, INF, INF, INF };

    // Each lane walks the col-point for its column l16; advance 16 points (48 floats)/tile.
    const float* gp = G + (size_t)l16 * 3u;

#pragma unroll 4
    for (int c0 = 0; c0 < C; c0 += 16) {
        float gx = gp[0], gy = gp[1], gz = gp[2];
        gp += 48;
        float g2 = gx * gx + gy * gy + gz * gz;

        // ---- B operand: 4x16 f32, columns = gt points, rows K = (gx, gy, gz, |g|^2).
        v2f b;
        b.x = half ? gz : gx;
        b.y = half ? g2 : gy;

        v8f c = {};   // zero accumulator
        // D[m][n] = -2 p_m . g_n + |g_n|^2
        v8f d = __builtin_amdgcn_wmma_f32_16x16x4_f32(
            false, a, false, b, (short)0, c, false, false);

#pragma unroll
        for (int r = 0; r < 8; ++r)
            mn[r] = fminf(mn[r], d[r]);
    }

    // ---- Reduce mins across the 16 lanes of each half (columns of the tile).
#pragma unroll
    for (int m = 1; m <= 8; m <<= 1) {
#pragma unroll
        for (int r = 0; r < 8; ++r)
            mn[r] = fminf(mn[r], __shfl_xor(mn[r], m, 32));
    }

    // Apply the monotone transform after the min: d_i = max(min_j D + |p_i|^2, 0).
    float s = 0.0f;
#pragma unroll
    for (int r = 0; r < 8; ++r)
        s += fmaxf(mn[r] + x2r[r], 0.0f);
    // lanes 0-15 hold sum over rows 0-7, lanes 16-31 over rows 8-15; combine halves.
    s += __shfl_xor(s, 16, 32);

    __shared__ float wsum[WAVES_PER_BLOCK];
    if (lane == 0) wsum[wave] = s;
    __syncthreads();
    if (tid == 0) {
        float t = 0.0f;
#pragma unroll
        for (int w = 0; w < WAVES_PER_BLOCK; ++w) t += wsum[w];
        partial[blockIdx.x] = t;
    }
}

// Deterministic final reduction: fixed-order tree, no float atomics.
__global__ __launch_bounds__(256) void chamfer_finalize(
    const float* __restrict__ p1, int n1, float scale1,
    const float* __restrict__ p2, int n2, float scale2,
    float* __restrict__ out)
{
    __shared__ float sm[256];
    int t = threadIdx.x;
    float s1 = 0.0f, s2 = 0.0f;
    for (int i = t; i < n1; i += 256) s1 += p1[i];
    for (int i = t; i < n2; i += 256) s2 += p2[i];
    sm[t] = s1 * scale1 + s2 * scale2;
    __syncthreads();
    for (int o = 128; o > 0; o >>= 1) {
        if (t < o) sm[t] += sm[t + o];
        __syncthreads();
    }
    if (t == 0) out[0] = sm[0];
}

extern "C" void kernel_launch(void* const* d_in, const int* in_sizes, int n_in,
                              void* d_out, int out_size, void* d_ws, size_t ws_size,
                              hipStream_t stream) {
    const float* pred = (const float*)d_in[0];   // [B, N, 3] fp32
    const float* gt   = (const float*)d_in[1];   // [B, M, 3] fp32
    float* out = (float*)d_out;
    float* wsf = (float*)d_ws;

    const int B = 4;
    const int N = in_sizes[0] / (3 * B);         // 8192
    const int M = in_sizes[1] / (3 * B);         // 8192

    const int bpbN = N / ROWS_PER_BLOCK;         // 64 row-blocks per batch
    const int bpbM = M / ROWS_PER_BLOCK;
    const int grid1 = B * bpbN;                  // 256
    const int grid2 = B * bpbM;                  // 256

    // Pass 1: rows = pred, cols = gt  -> sum_i min_j d  (dist1 numerator)
    chamfer_min_rows<<<grid1, 256, 0, stream>>>(pred, gt, N, M, bpbN, wsf);
    // Pass 2: rows = gt, cols = pred -> sum_j min_i d  (dist2 numerator)
    chamfer_min_rows<<<grid2, 256, 0, stream>>>(gt, pred, M, N, bpbM, wsf + grid1);

    chamfer_finalize<<<1, 256, 0, stream>>>(
        wsf, grid1, 1.0f / (float)(B * N),
        wsf + grid1, grid2, 1.0f / (float)(B * M),
        out);
}